// MultipleGuidos_70145405878966
// MI455X (gfx1250) — compile-verified
//
#include <hip/hip_runtime.h>
#include <hip/hip_bf16.h>

// Per-row 4-oscillator Kuramoto RK4 step + softmax head.
// Exploits: C symmetric + sin odd -> 6 pairs per f-eval; sin(2d)=2*s*c -> no
// extra transcendentals for the anti-phase term (12 TRANS per f-eval, 53 total
// per row). v_rcp_f32 instead of IEEE div; softmax needs no max-subtraction
// since logits are bounded in [0, 2/sqrt(pi)].

__device__ __forceinline__ float fast_rcp(float x) {
    return __builtin_amdgcn_rcpf(x);
}

// g_ij = w_ij * (sin(d) + inv_k * sin(2d)) = (w*s) * fma(2*inv_k, c, 1)
__device__ __forceinline__ void eval_f(const float ph0, const float ph1,
                                       const float ph2, const float ph3,
                                       const float fq0, const float fq1,
                                       const float fq2, const float fq3,
                                       const float x0,  const float x1,
                                       const float w01, const float w02,
                                       const float w03, const float w12,
                                       const float w13, const float w23,
                                       const float ik2,   // 2 / k_param
                                       float& f0, float& f1, float& f2, float& f3) {
    float d, s, c;
    d = ph0 - ph1; s = __sinf(d); c = __cosf(d);
    const float g01 = (w01 * s) * __builtin_fmaf(ik2, c, 1.0f);
    d = ph0 - ph2; s = __sinf(d); c = __cosf(d);
    const float g02 = (w02 * s) * __builtin_fmaf(ik2, c, 1.0f);
    d = ph0 - ph3; s = __sinf(d); c = __cosf(d);
    const float g03 = (w03 * s) * __builtin_fmaf(ik2, c, 1.0f);
    d = ph1 - ph2; s = __sinf(d); c = __cosf(d);
    const float g12 = (w12 * s) * __builtin_fmaf(ik2, c, 1.0f);
    d = ph1 - ph3; s = __sinf(d); c = __cosf(d);
    const float g13 = (w13 * s) * __builtin_fmaf(ik2, c, 1.0f);
    d = ph2 - ph3; s = __sinf(d); c = __cosf(d);
    const float g23 = (w23 * s) * __builtin_fmaf(ik2, c, 1.0f);

    f0 = fq0 + x0 - (g01 + g02 + g03);
    f1 = fq1 + x1 + g01 - (g12 + g13);
    f2 = fq2       + g02 + g12 - g23;
    f3 = fq3       + g03 + g13 + g23;
}

__global__ __launch_bounds__(256) void MultipleGuidos_70145405878966_kernel(
    const float* __restrict__ inputs,     // (N,2)
    const float* __restrict__ phases,     // (N,4)
    const float* __restrict__ frequency,  // (N,2)
    const float* __restrict__ pcoup,      // (N,4)
    const float* __restrict__ kparam,     // (N,)
    const int*   __restrict__ fs_p,       // scalar
    float*       __restrict__ out,        // (N,3)
    int n) {
    const int i = blockIdx.x * blockDim.x + threadIdx.x;
    if (i >= n) return;

    const float dt  = fast_rcp((float)fs_p[0]);   // uniform
    const float hdt = 0.5f * dt;                  // RK4 half-step
    const float sdt = dt * (1.0f / 6.0f);         // dphi scale

    // Vectorized loads: b64 + b128 + b64 + b128 + b32 per lane.
    const float2 xin = reinterpret_cast<const float2*>(inputs)[i];
    const float4 ph  = reinterpret_cast<const float4*>(phases)[i];
    const float2 fr  = reinterpret_cast<const float2*>(frequency)[i];
    const float4 pc  = reinterpret_cast<const float4*>(pcoup)[i];
    const float  kv  = kparam[i];

    const float ik2 = 2.0f * fast_rcp(kv);
    // Coupling weights (symmetric matrix, upper triangle):
    const float w01 = pc.x;  // a_s
    const float w02 = pc.y;  // a_i
    const float w03 = pc.z;  // a_c
    const float w12 = pc.z;  // a_c
    const float w13 = pc.y;  // a_i
    const float w23 = pc.w;  // a_m

    const float fq0 = fr.x, fq1 = fr.x, fq2 = fr.y, fq3 = fr.y;
    const float x0 = xin.x, x1 = xin.y;

    float f1_0, f1_1, f1_2, f1_3;   // stage derivatives (un-scaled by dt)
    float f2_0, f2_1, f2_2, f2_3;
    float f3_0, f3_1, f3_2, f3_3;
    float f4_0, f4_1, f4_2, f4_3;

    // stage 1: f(ph)
    eval_f(ph.x, ph.y, ph.z, ph.w, fq0, fq1, fq2, fq3, x0, x1,
           w01, w02, w03, w12, w13, w23, ik2, f1_0, f1_1, f1_2, f1_3);

    // stage 2: f(ph + 0.5*dt*f1)   -- single fma per component
    eval_f(__builtin_fmaf(hdt, f1_0, ph.x), __builtin_fmaf(hdt, f1_1, ph.y),
           __builtin_fmaf(hdt, f1_2, ph.z), __builtin_fmaf(hdt, f1_3, ph.w),
           fq0, fq1, fq2, fq3, x0, x1,
           w01, w02, w03, w12, w13, w23, ik2, f2_0, f2_1, f2_2, f2_3);

    // stage 3: f(ph + 0.5*dt*f2)
    eval_f(__builtin_fmaf(hdt, f2_0, ph.x), __builtin_fmaf(hdt, f2_1, ph.y),
           __builtin_fmaf(hdt, f2_2, ph.z), __builtin_fmaf(hdt, f2_3, ph.w),
           fq0, fq1, fq2, fq3, x0, x1,
           w01, w02, w03, w12, w13, w23, ik2, f3_0, f3_1, f3_2, f3_3);

    // stage 4: f(ph + dt*f3)
    eval_f(__builtin_fmaf(dt, f3_0, ph.x), __builtin_fmaf(dt, f3_1, ph.y),
           __builtin_fmaf(dt, f3_2, ph.z), __builtin_fmaf(dt, f3_3, ph.w),
           fq0, fq1, fq2, fq3, x0, x1,
           w01, w02, w03, w12, w13, w23, ik2, f4_0, f4_1, f4_2, f4_3);

    // new_phases[2], new_phases[3]: ph + dt/6 * (f1 + 2 f2 + 2 f3 + f4)
    const float np2 = __builtin_fmaf(
        sdt, f1_2 + 2.0f * f2_2 + 2.0f * f3_2 + f4_2, ph.z);
    const float np3 = __builtin_fmaf(
        sdt, f1_3 + 2.0f * f2_3 + 2.0f * f3_3 + f4_3, ph.w);

    // theta = new_phases[3] - new_phases[2]
    const float theta = np3 - np2;
    const float st = __sinf(theta);
    const float ct = __cosf(theta);
    const float a = 0.5641895835477563f;  // sqrt(1/pi)
    // [a*(1 - sin(-t)), a*(1 - sin(t)), a*(1 - cos(pi - t))]
    const float p0 = a * (1.0f + st);
    const float p1 = a * (1.0f - st);
    const float p2 = a * (1.0f + ct);

    // softmax; logits bounded in [0, 1.13] so no max-shift needed
    const float e0 = __expf(p0);
    const float e1 = __expf(p1);
    const float e2 = __expf(p2);
    const float inv_sum = fast_rcp(e0 + e1 + e2);

    float* o = out + 3 * (size_t)i;
    o[0] = e0 * inv_sum;
    o[1] = e1 * inv_sum;
    o[2] = e2 * inv_sum;
}

extern "C" void kernel_launch(void* const* d_in, const int* in_sizes, int n_in,
                              void* d_out, int out_size, void* d_ws, size_t ws_size,
                              hipStream_t stream) {
    const float* inputs    = (const float*)d_in[0];  // (N,2)
    const float* phases    = (const float*)d_in[1];  // (N,4)
    const float* frequency = (const float*)d_in[2];  // (N,2)
    const float* pcoup     = (const float*)d_in[3];  // (N,4)
    const float* kparam    = (const float*)d_in[4];  // (N,)
    const int*   fs_p      = (const int*)d_in[5];    // scalar 100

    const int n = in_sizes[4];  // k_param element count == N
    float* out = (float*)d_out;

    const int block = 256;  // 8 wave32s
    const int grid  = (n + block - 1) / block;
    MultipleGuidos_70145405878966_kernel<<<grid, block, 0, stream>>>(
        inputs, phases, frequency, pcoup, kparam, fs_p, out, n);
}